// PseudoMambaBlock_4844723109968
// MI455X (gfx1250) — compile-verified
//
#include <hip/hip_runtime.h>
#include <hip/hip_bf16.h>

typedef __attribute__((ext_vector_type(2))) float v2f;
typedef __attribute__((ext_vector_type(4))) float v4f;
typedef __attribute__((ext_vector_type(8))) float v8f;
typedef __attribute__((ext_vector_type(4))) int   v4i;

#define SEQLEN   2048
#define DMODEL   1024
#define DINNER   2048
#define DSTATE   16

// GEMM macro-tile config
#define BM  128
#define BN  64
#define KC  32
#define LDT 36           // padded LDS row stride (floats): 144B = 16B-aligned,
                         // 36r mod 64 hits each multiple-of-4 bank once -> conflict-free

__device__ __forceinline__ float silu_f(float v) {
    return v / (1.0f + __expf(-v));
}
__device__ __forceinline__ float softplus_f(float v) {
    return (v > 20.0f) ? v : log1pf(__expf(v));
}

// ---------------------------------------------------------------------------
// CDNA5 async global->LDS copy (ASYNCcnt path), with sync fallback.
// Builtin prototype (per hipcc diagnostic): (int4* gsrc, int4* ldsdst, Ii, Ii)
// ---------------------------------------------------------------------------
#if defined(__has_builtin)
#if __has_builtin(__builtin_amdgcn_global_load_async_to_lds_b128)
#define USE_ASYNC_LDS 1
#endif
#endif

#ifdef USE_ASYNC_LDS
__device__ __forceinline__ void cp_async_b128(float* lds_dst, const float* g_src) {
    __builtin_amdgcn_global_load_async_to_lds_b128(
        (v4i*)(g_src), (v4i*)(lds_dst), 0, 0);
}
__device__ __forceinline__ void wait_async0() {
#if __has_builtin(__builtin_amdgcn_s_wait_asynccnt)
    __builtin_amdgcn_s_wait_asynccnt(0);
#else
    asm volatile("s_wait_asynccnt 0x0" ::: "memory");
#endif
}
#endif

// ---------------------------------------------------------------------------
// Pipelined fp32 WMMA GEMM:  C[M,N] = A[M,K] * B[N,K]^T  (both K-contiguous)
// Block = 8 waves -> 128x64 macro-tile; Kc=32 double-buffered through LDS.
// Async path: global_load_async_to_lds_b128 + s_wait_asynccnt (1 wait+1 barrier
// per K-slab). Fallback: register-staged sync pipeline.
// EPI: 0 = none, 1 = softplus(C + aux[n]), 2 = C + aux[m*N+n] (residual)
// ---------------------------------------------------------------------------
template<int EPI>
__global__ __launch_bounds__(256) void wmma_gemm_tiled(
    const float* __restrict__ A, const float* __restrict__ B,
    const float* __restrict__ aux, float* __restrict__ C,
    int M, int N, int K)
{
    __shared__ float sA[2][BM * LDT];   // 2 * 128*36*4 = 36.9 KB
    __shared__ float sB[2][BN * LDT];   // 2 *  64*36*4 = 18.4 KB

    const int tid      = threadIdx.x;
    const int wave     = tid >> 5;
    const int lane     = tid & 31;
    const int lane16   = lane & 15;
    const int laneHalf = lane >> 4;       // 0: K=0,1 / M=i   1: K=2,3 / M=8+i
    const int kOff     = laneHalf * 2;

    const int tilesN = N / BN;
    const int tm = blockIdx.x / tilesN;
    const int tn = blockIdx.x % tilesN;

    // cooperative tile-load mapping: one float4 per (row, c4) slot
    const int ldRow = tid >> 3;            // 0..31
    const int ldC4  = (tid & 7) * 4;       // 0..28

    const float* gA = A + (size_t)(tm * BM + ldRow) * K + ldC4;
    const float* gB = B + (size_t)(tn * BN + ldRow) * K + ldC4;

    v8f acc[4] = {};

    const int KT = K / KC;

    auto compute = [&](int buf) {
        const float* aBase = &sA[buf][(wave * 16 + lane16) * LDT + kOff];
        const float* bBase = &sB[buf][lane16 * LDT + kOff];
#pragma unroll
        for (int kk = 0; kk < KC / 4; ++kk) {
            v2f a = *(const v2f*)(aBase + kk * 4);
#pragma unroll
            for (int j = 0; j < 4; ++j) {
                v2f b = *(const v2f*)(bBase + j * 16 * LDT + kk * 4);
                acc[j] = __builtin_amdgcn_wmma_f32_16x16x4_f32(
                    false, a, false, b, (short)0, acc[j], false, false);
            }
        }
    };

#ifdef USE_ASYNC_LDS
    auto issue = [&](int kt, int buf) {
        const size_t k0 = (size_t)kt * KC;
#pragma unroll
        for (int it = 0; it < 4; ++it)     // A: 128 rows
            cp_async_b128(&sA[buf][(ldRow + it * 32) * LDT + ldC4],
                          gA + (size_t)(it * 32) * K + k0);
#pragma unroll
        for (int it = 0; it < 2; ++it)     // B: 64 rows
            cp_async_b128(&sB[buf][(ldRow + it * 32) * LDT + ldC4],
                          gB + (size_t)(it * 32) * K + k0);
    };

    issue(0, 0);
    for (int kt = 0; kt < KT; ++kt) {
        const int buf = kt & 1;
        wait_async0();                     // this wave's copies into buf landed
        __syncthreads();                   // everyone's copies landed / buf^1 free
        if (kt + 1 < KT) issue(kt + 1, buf ^ 1);
        compute(buf);
    }
#else
    v4f rA[4], rB[2];
    auto fetch = [&](int kt) {
        const size_t k0 = (size_t)kt * KC;
#pragma unroll
        for (int it = 0; it < 4; ++it)
            rA[it] = *(const v4f*)(gA + (size_t)(it * 32) * K + k0);
#pragma unroll
        for (int it = 0; it < 2; ++it)
            rB[it] = *(const v4f*)(gB + (size_t)(it * 32) * K + k0);
    };
    auto stage = [&](int buf) {
#pragma unroll
        for (int it = 0; it < 4; ++it)
            *(v4f*)&sA[buf][(ldRow + it * 32) * LDT + ldC4] = rA[it];
#pragma unroll
        for (int it = 0; it < 2; ++it)
            *(v4f*)&sB[buf][(ldRow + it * 32) * LDT + ldC4] = rB[it];
    };

    fetch(0);
    stage(0);
    __syncthreads();
    for (int kt = 0; kt < KT; ++kt) {
        if (kt + 1 < KT) fetch(kt + 1);    // global loads overlap compute below
        compute(kt & 1);
        __syncthreads();
        if (kt + 1 < KT) stage((kt + 1) & 1);
        __syncthreads();
    }
#endif

    // C/D layout: VGPR i -> M = i (lanes 0-15) or M = 8+i (lanes 16-31); N = lane16
#pragma unroll
    for (int j = 0; j < 4; ++j) {
#pragma unroll
        for (int i = 0; i < 8; ++i) {
            int m = tm * BM + wave * 16 + i + laneHalf * 8;
            int n = tn * BN + j * 16 + lane16;
            float v = acc[j][i];
            if (EPI == 1)      v = softplus_f(v + aux[n]);
            else if (EPI == 2) v = v + aux[(size_t)m * N + n];
            C[(size_t)m * N + n] = v;
        }
    }
}

// ---------------------------------------------------------------------------
// Direct (non-tiled) WMMA GEMM for the tiny N=32 projection (W_x).
// ---------------------------------------------------------------------------
template<int NT>
__global__ __launch_bounds__(256) void wmma_gemm_small(
    const float* __restrict__ A, const float* __restrict__ B,
    float* __restrict__ C, int M, int N, int K)
{
    const int wave     = threadIdx.x >> 5;
    const int lane     = threadIdx.x & 31;
    const int tiles_n  = N / (16 * NT);
    const int tileIdx  = blockIdx.x * 8 + wave;
    const int tm       = tileIdx / tiles_n;
    const int tn       = tileIdx % tiles_n;
    const int laneHalf = lane >> 4;
    const int lane16   = lane & 15;
    const int kOff     = laneHalf * 2;

    const float* Arow = A + (size_t)(tm * 16 + lane16) * K + kOff;
    const float* Brow[NT];
#pragma unroll
    for (int j = 0; j < NT; ++j)
        Brow[j] = B + (size_t)((tn * NT + j) * 16 + lane16) * K + kOff;

    v8f acc[NT] = {};
    for (int k = 0; k < K; k += 4) {
        v2f a = *(const v2f*)(Arow + k);
#pragma unroll
        for (int j = 0; j < NT; ++j) {
            v2f b = *(const v2f*)(Brow[j] + k);
            acc[j] = __builtin_amdgcn_wmma_f32_16x16x4_f32(
                false, a, false, b, (short)0, acc[j], false, false);
        }
    }
#pragma unroll
    for (int j = 0; j < NT; ++j)
#pragma unroll
        for (int i = 0; i < 8; ++i) {
            int m = tm * 16 + i + laneHalf * 8;
            int n = (tn * NT + j) * 16 + lane16;
            C[(size_t)m * N + n] = acc[j][i];
        }
}

// ---------------------------------------------------------------------------
// Causal depthwise conv1d (k=4, pad-left 3) + bias + SiLU.
// ---------------------------------------------------------------------------
__global__ __launch_bounds__(256) void conv_silu_kernel(
    const float* __restrict__ xz, const float* __restrict__ conv_w,
    const float* __restrict__ conv_b, float* __restrict__ x_ssm)
{
    int t = blockIdx.x * 256 + threadIdx.x;       // t = l*DINNER + e
    int l = t >> 11;
    int e = t & (DINNER - 1);
    float acc = conv_b[e];
#pragma unroll
    for (int j = 0; j < 4; ++j) {
        int li = l - 3 + j;
        float xv = (li >= 0) ? xz[(size_t)li * (2 * DINNER) + e] : 0.0f;
        acc = fmaf(conv_w[e * 4 + j], xv, acc);
    }
    x_ssm[t] = silu_f(acc);
}

// ---------------------------------------------------------------------------
// Selective scan: one thread per channel e; h[16]/A[16] in registers; B/C
// staged through LDS in 128-step chunks. Fuses +D*x_ssm and *silu(z).
// ---------------------------------------------------------------------------
__global__ __launch_bounds__(256) void scan_kernel(
    const float* __restrict__ dt,    const float* __restrict__ x_ssm,
    const float* __restrict__ x_dbl, const float* __restrict__ xz,
    const float* __restrict__ A_log, const float* __restrict__ D_param,
    float* __restrict__ y)
{
    __shared__ float bc[128 * 32];
    const int e = blockIdx.x * 256 + threadIdx.x;

    float Ae[DSTATE], h[DSTATE];
#pragma unroll
    for (int s = 0; s < DSTATE; ++s) {
        Ae[s] = -__expf(A_log[e * DSTATE + s]);
        h[s]  = 0.0f;
    }
    const float De = D_param[e];

    for (int l0 = 0; l0 < SEQLEN; l0 += 128) {
        __syncthreads();
#pragma unroll
        for (int i = 0; i < 16; ++i)
            bc[threadIdx.x + i * 256] = x_dbl[(size_t)l0 * 32 + threadIdx.x + i * 256];
        __syncthreads();

        for (int ls = 0; ls < 128; ++ls) {
            const int l     = l0 + ls;
            const float dtv = dt[(size_t)l * DINNER + e];
            const float xv  = x_ssm[(size_t)l * DINNER + e];
            const float dtx = dtv * xv;
            const float* Bv = &bc[ls * 32];
            float yt = 0.0f;
#pragma unroll
            for (int s = 0; s < DSTATE; ++s) {
                float dA = __expf(dtv * Ae[s]);
                h[s] = fmaf(dA, h[s], dtx * Bv[s]);
                yt   = fmaf(h[s], Bv[16 + s], yt);
            }
            const float zv = xz[(size_t)l * (2 * DINNER) + DINNER + e];
            y[(size_t)l * DINNER + e] = (yt + De * xv) * silu_f(zv);
        }
    }
}

// ---------------------------------------------------------------------------
// Launch: x, W_in, conv_w, conv_b, A_log, D_param, W_x, W_dt, b_dt, W_out
// ---------------------------------------------------------------------------
extern "C" void kernel_launch(void* const* d_in, const int* in_sizes, int n_in,
                              void* d_out, int out_size, void* d_ws, size_t ws_size,
                              hipStream_t stream) {
    const float* x      = (const float*)d_in[0];
    const float* W_in   = (const float*)d_in[1];
    const float* conv_w = (const float*)d_in[2];
    const float* conv_b = (const float*)d_in[3];
    const float* A_log  = (const float*)d_in[4];
    const float* D_par  = (const float*)d_in[5];
    const float* W_x    = (const float*)d_in[6];
    const float* W_dt   = (const float*)d_in[7];
    const float* b_dt   = (const float*)d_in[8];
    const float* W_out  = (const float*)d_in[9];
    float* out = (float*)d_out;

    float* ws    = (float*)d_ws;
    float* xz    = ws;                                   // 2048*4096
    float* x_ssm = xz    + (size_t)SEQLEN * 2 * DINNER;  // 2048*2048
    float* dt    = x_ssm + (size_t)SEQLEN * DINNER;      // 2048*2048
    float* x_dbl = dt    + (size_t)SEQLEN * DINNER;      // 2048*32
    float* y     = x_dbl + (size_t)SEQLEN * 2 * DSTATE;  // 2048*2048

    // xz = x @ W_in^T : M=2048 N=4096 K=1024 -> (2048/128)*(4096/64) = 1024 blocks
    wmma_gemm_tiled<0><<<1024, 256, 0, stream>>>(x, W_in, nullptr, xz,
                                                 SEQLEN, 2 * DINNER, DMODEL);
    // conv + silu -> x_ssm
    conv_silu_kernel<<<(SEQLEN * DINNER) / 256, 256, 0, stream>>>(xz, conv_w, conv_b, x_ssm);
    // dt = softplus(x_ssm @ W_dt^T + b_dt) : 16*32 = 512 blocks
    wmma_gemm_tiled<1><<<512, 256, 0, stream>>>(x_ssm, W_dt, b_dt, dt,
                                                SEQLEN, DINNER, DINNER);
    // x_dbl = x_ssm @ W_x^T : M=2048 N=32 K=2048 (tiny) -> direct WMMA kernel
    wmma_gemm_small<2><<<16, 256, 0, stream>>>(x_ssm, W_x, x_dbl,
                                               SEQLEN, 2 * DSTATE, DINNER);
    // selective scan (fuses +D*x_ssm and *silu(z))
    scan_kernel<<<DINNER / 256, 256, 0, stream>>>(dt, x_ssm, x_dbl, xz, A_log, D_par, y);
    // out = y @ W_out^T + x : 16*16 = 256 blocks
    wmma_gemm_tiled<2><<<256, 256, 0, stream>>>(y, W_out, x, out,
                                                SEQLEN, DMODEL, DINNER);
}